// ResContextBlock_15968688407222
// MI455X (gfx1250) — compile-verified
//
#include <hip/hip_runtime.h>

#define D_   480
#define H_   360
#define W_   32
#define B_   2
#define COUT 32
#define EPSF 1e-5f
#define NEGF 0.01f
#define NBR_STRIDE 12   // 9 taps padded to 12 ints -> 48B rows, 16B aligned
#define TPW  4          // tiles (of 16 points) per wave

typedef __attribute__((ext_vector_type(2))) float v2f;
typedef __attribute__((ext_vector_type(8))) float v8f;

// ---------------- grid build ----------------
__global__ void grid_init_kernel(int4* __restrict__ grid, int n4) {
    int i = blockIdx.x * blockDim.x + threadIdx.x;
    if (i < n4) grid[i] = make_int4(-1, -1, -1, -1);
}

__global__ void zrow_init_kernel(float* __restrict__ z) {
    z[threadIdx.x] = 0.0f;   // <<<1,32>>>
}

__global__ void grid_scatter_kernel(const int* __restrict__ coords,
                                    int* __restrict__ grid, int n) {
    int i = blockIdx.x * blockDim.x + threadIdx.x;
    if (i >= n) return;
    int b = coords[4 * i + 0];
    int z = coords[4 * i + 1];
    int y = coords[4 * i + 2];
    int x = coords[4 * i + 3];
    grid[((b * D_ + z) * H_ + y) * W_ + x] = i;
}

// ---------------- neighbor tables (shared by 2 layers each) ----------------
__global__ void nbr_build_kernel(const int* __restrict__ coords,
                                 const int* __restrict__ grid,
                                 int* __restrict__ nbr133,
                                 int* __restrict__ nbr313, int n) {
    int i = blockIdx.x * blockDim.x + threadIdx.x;
    if (i >= n) return;
    int b = coords[4 * i + 0];
    int z = coords[4 * i + 1];
    int y = coords[4 * i + 2];
    int x = coords[4 * i + 3];
    // OFF_133: (0, dy, dx), dy outer, dx inner
    int k = 0;
    #pragma unroll
    for (int dy = -1; dy <= 1; ++dy) {
        #pragma unroll
        for (int dx = -1; dx <= 1; ++dx) {
            int ny = y + dy, nx = x + dx;
            int v = -1;
            if (ny >= 0 && ny < H_ && nx >= 0 && nx < W_)
                v = grid[((b * D_ + z) * H_ + ny) * W_ + nx];
            nbr133[i * NBR_STRIDE + k] = v;
            ++k;
        }
    }
    // OFF_313: (dz, 0, dx), dz outer, dx inner
    k = 0;
    #pragma unroll
    for (int dz = -1; dz <= 1; ++dz) {
        #pragma unroll
        for (int dx = -1; dx <= 1; ++dx) {
            int nz = z + dz, nx = x + dx;
            int v = -1;
            if (nz >= 0 && nz < D_ && nx >= 0 && nx < W_)
                v = grid[((b * D_ + nz) * H_ + y) * W_ + nx];
            nbr313[i * NBR_STRIDE + k] = v;
            ++k;
        }
    }
}

// ---------------- fused gather + WMMA GEMM + BN + LeakyReLU ----------------
// One wave32 handles TPW tiles of 16 points; each tile -> 16x32 outputs in two
// f32 WMMA accumulators. LDS weight staging + BN constants amortized over TPW.
// A (16x4 f32): lane<16 holds A[row][c+0..1], lane>=16 holds A[row][c+2..3]
// B (4x16 f32): vgpr j, lane l<16 -> B[j][l]; l>=16 -> B[2+j][l-16]
// C/D (16x16 f32): lane l, vgpr v -> (M = v + 8*(l>=16), N = l&15)
//
// Weights staged into LDS PRE-SWIZZLED as pairs
//   Wl[((k*(CIN/2) + ci/2)*COUT + co)*2 + (ci&1)] = W[k][ci][co]
// so each B fragment is one aligned ds_load_b64 (2addr-fusable, no repacking).
// A-row base pointers are selected branchlessly against a zero row so all
// global gathers are unconditional and get s_clause-batched by the scheduler.
template <int CIN, bool ADD_RES>
__global__ void __launch_bounds__(256)
conv_bn_kernel(const float* __restrict__ feats,   // n x CIN
               const float* __restrict__ Wg,      // 9 x CIN x 32
               const int*   __restrict__ nbr,     // n x NBR_STRIDE (9 used)
               const float* __restrict__ zrow,    // >= CIN zero floats
               const float* __restrict__ gamma,
               const float* __restrict__ beta,
               const float* __restrict__ mean,
               const float* __restrict__ var,
               const float* __restrict__ res,     // shortcut (may be null)
               float* __restrict__ out,           // n x 32
               int n) {
    __shared__ float Wl[9 * CIN * COUT];
    for (int i = threadIdx.x; i < 9 * CIN * COUT; i += blockDim.x) {
        int k  = i / (CIN * COUT);
        int r  = i - k * (CIN * COUT);
        int ci = r / COUT;
        int co = r - ci * COUT;
        Wl[((k * (CIN / 2) + (ci >> 1)) * COUT + co) * 2 + (ci & 1)] = Wg[i];
    }
    __syncthreads();

    const int lane = threadIdx.x & 31;
    const int wv   = threadIdx.x >> 5;
    const int row  = lane & 15;                 // point-in-tile for A; col for B/D
    const int half = lane >> 4;                 // K-half select

    // BN constants depend only on this lane's output channels: hoist per wave.
    float scl[2], bia[2];
    #pragma unroll
    for (int t2 = 0; t2 < 2; ++t2) {
        const int co = t2 * 16 + row;
        scl[t2] = gamma[co] * rsqrtf(var[co] + EPSF);
        bia[t2] = beta[co] - mean[co] * scl[t2];
    }

    const int tile0 = (blockIdx.x * (blockDim.x >> 5) + wv) * TPW;

    #pragma unroll 1
    for (int t = 0; t < TPW; ++t) {
        const int tile = tile0 + t;
        if (tile * 16 >= n) break;              // uniform per wave

        const int point  = tile * 16 + row;
        const int pclamp = point < n ? point : n - 1;

        // all 9 taps in 3 vector loads (rows are 16B aligned: 48B stride)
        const int4* np = (const int4*)(nbr + (size_t)pclamp * NBR_STRIDE);
        const int4 n03 = np[0];
        const int4 n47 = np[1];
        const int  n8  = ((const int*)(np + 2))[0];
        const int nbrs[9] = {n03.x, n03.y, n03.z, n03.w,
                             n47.x, n47.y, n47.z, n47.w, n8};

        v8f acc0 = {};
        v8f acc1 = {};

        #pragma unroll
        for (int k = 0; k < 9; ++k) {
            const int nb = nbrs[k];
            const float* frow = (nb >= 0) ? (feats + (size_t)nb * CIN) : zrow;
            #pragma unroll
            for (int c = 0; c < CIN; c += 4) {
                const v2f A = *(const v2f*)(frow + c + 2 * half);
                const float* wp =
                    &Wl[((k * (CIN / 2) + (c >> 1) + half) * COUT + row) * 2];
                const v2f B0 = *(const v2f*)(wp);        // col-tile 0
                const v2f B1 = *(const v2f*)(wp + 32);   // col-tile 1 (+16 co)
                acc0 = __builtin_amdgcn_wmma_f32_16x16x4_f32(
                    false, A, false, B0, (short)0, acc0, false, false);
                acc1 = __builtin_amdgcn_wmma_f32_16x16x4_f32(
                    false, A, false, B1, (short)0, acc1, false, false);
            }
        }

        // BN + LeakyReLU + (optional residual) + store
        #pragma unroll
        for (int t2 = 0; t2 < 2; ++t2) {
            const int co = t2 * 16 + row;
            v8f acc = t2 ? acc1 : acc0;
            #pragma unroll
            for (int v = 0; v < 8; ++v) {
                float yv = acc[v] * scl[t2] + bia[t2];
                yv = (yv >= 0.0f) ? yv : NEGF * yv;
                const int orow = tile * 16 + half * 8 + v;
                if (orow < n) {
                    const int oidx = orow * COUT + co;
                    if (ADD_RES) yv += res[oidx];
                    out[oidx] = yv;
                }
            }
        }
    }
}

// ---------------- launcher ----------------
extern "C" void kernel_launch(void* const* d_in, const int* in_sizes, int n_in,
                              void* d_out, int out_size, void* d_ws, size_t ws_size,
                              hipStream_t stream) {
    const float* feats    = (const float*)d_in[0];
    const int*   coords   = (const int*)  d_in[1];
    const float* W1       = (const float*)d_in[2];
    const float* W2       = (const float*)d_in[3];
    const float* W3       = (const float*)d_in[4];
    const float* W4       = (const float*)d_in[5];
    const float* bn_gamma = (const float*)d_in[6];
    const float* bn_beta  = (const float*)d_in[7];
    const float* bn_mean  = (const float*)d_in[8];
    const float* bn_var   = (const float*)d_in[9];
    float* out = (float*)d_out;

    const int n = in_sizes[0] / 16;   // feats is N x CIN(=16)

    // workspace layout (~115 MB): grid | nbr133 | nbr313 | bufA | bufB | zrow
    const size_t GRID_ELEMS = (size_t)B_ * D_ * H_ * W_;   // 11,059,200
    int*   grid   = (int*)d_ws;
    int*   nbr133 = grid + GRID_ELEMS;
    int*   nbr313 = nbr133 + (size_t)n * NBR_STRIDE;
    float* bufA   = (float*)(nbr313 + (size_t)n * NBR_STRIDE);
    float* bufB   = bufA + (size_t)n * COUT;
    float* zrow   = bufB + (size_t)n * COUT;

    // 1) dense grid build (44MB -> resident in 192MB L2 during conv passes)
    const int n4 = (int)(GRID_ELEMS / 4);
    grid_init_kernel<<<(n4 + 255) / 256, 256, 0, stream>>>((int4*)grid, n4);
    zrow_init_kernel<<<1, 32, 0, stream>>>(zrow);
    grid_scatter_kernel<<<(n + 255) / 256, 256, 0, stream>>>(coords, grid, n);

    // 2) neighbor tables (each reused by 2 layers)
    nbr_build_kernel<<<(n + 255) / 256, 256, 0, stream>>>(coords, grid, nbr133, nbr313, n);

    // 3) four fused conv+BN+LReLU passes
    const int tiles  = (n + 15) / 16;
    const int wavesN = (tiles + TPW - 1) / TPW;
    const int blocks = (wavesN + 7) / 8;   // 8 waves per 256-thread block

    conv_bn_kernel<16, false><<<blocks, 256, 0, stream>>>(
        feats, W1, nbr133, zrow,
        bn_gamma + 0,  bn_beta + 0,  bn_mean + 0,  bn_var + 0,
        nullptr, bufA, n);
    conv_bn_kernel<32, false><<<blocks, 256, 0, stream>>>(
        bufA,  W2, nbr313, zrow,
        bn_gamma + 32, bn_beta + 32, bn_mean + 32, bn_var + 32,
        nullptr, bufB, n);                                   // shortcut
    conv_bn_kernel<16, false><<<blocks, 256, 0, stream>>>(
        feats, W3, nbr313, zrow,
        bn_gamma + 64, bn_beta + 64, bn_mean + 64, bn_var + 64,
        nullptr, bufA, n);
    conv_bn_kernel<32, true><<<blocks, 256, 0, stream>>>(
        bufA,  W4, nbr133, zrow,
        bn_gamma + 96, bn_beta + 96, bn_mean + 96, bn_var + 96,
        bufB, out, n);                                       // resA + shortcut
}